// Post_Prob_24498493456605
// MI455X (gfx1250) — compile-verified
//
#include <hip/hip_runtime.h>
#include <hip/hip_bf16.h>

typedef float v2f __attribute__((ext_vector_type(2)));
typedef float v4f __attribute__((ext_vector_type(4)));
typedef float v8f __attribute__((ext_vector_type(8)));

#define NPTS   512
#define NC     64
#define CELLS  (NC * NC)      // 4096
#define CHUNK  64
#define NCH    (NPTS / CHUNK) // 8
#define NROWS  513
#define INV2S2 0.0078125f     // 1 / (2 * 8^2)

// ---------------------------------------------------------------------------
// Stage 1: per batch -- exp factors Ey/Ex, S = Ey^T * Ex via v_wmma (fp32,
// 16x16x4, K=512), exact min-distance sweep, background row + 1/denom.
// 32 blocks (one per batch) x 512 threads (16 waves -> 16 WMMA tiles).
// LDS: 4 * 64*64*4B = 64 KB.
// ---------------------------------------------------------------------------
__global__ __launch_bounds__(512)
void post_prob_stage1(const float* __restrict__ points,
                      const float* __restrict__ st_sizes,
                      const float* __restrict__ cood,
                      float* __restrict__ wsEy,
                      float* __restrict__ wsEx,
                      float* __restrict__ wsInv,
                      float* __restrict__ out) {
    __shared__ float sEy[CHUNK][NC];
    __shared__ float sEx[CHUNK][NC];
    __shared__ float sDy[CHUNK][NC];
    __shared__ float sDx[CHUNK][NC];

    const int b    = blockIdx.x;
    const int tid  = threadIdx.x;
    const int lane = tid & 31;
    const int wave = tid >> 5;       // 0..15
    const int mi   = lane & 15;      // row/col within tile
    const int kh   = lane >> 4;      // K half-select per ISA f32 A-layout
    const int ty   = wave >> 2;      // tile row   (0..3)
    const int tx   = wave & 3;       // tile col   (0..3)

    const float* pts = points + (size_t)b * NPTS * 2;

    float minv[8];
#pragma unroll
    for (int j = 0; j < 8; ++j) minv[j] = 3.4e38f;

    v8f acc;
#pragma unroll
    for (int v = 0; v < 8; ++v) acc[v] = 0.0f;

    for (int ch = 0; ch < NCH; ++ch) {
        // ---- fill chunk: squared 1-D distances + exp factors -------------
        for (int idx = tid; idx < CHUNK * NC; idx += 512) {   // 8 iters, uniform
            const int n  = idx >> 6;
            const int cc = idx & 63;
            const int ng = ch * CHUNK + n;
            const float px = pts[2 * ng + 0];
            const float py = pts[2 * ng + 1];
            const float c  = cood[cc];
            const float dx = px - c;
            const float dy = py - c;
            const float dx2 = dx * dx;
            const float dy2 = dy * dy;
            const float ex = __expf(-dx2 * INV2S2);
            const float ey = __expf(-dy2 * INV2S2);
            sDx[n][cc] = dx2;
            sDy[n][cc] = dy2;
            sEx[n][cc] = ex;
            sEy[n][cc] = ey;
            wsEx[((size_t)b * NPTS + ng) * NC + cc] = ex;
            wsEy[((size_t)b * NPTS + ng) * NC + cc] = ey;
        }
        __syncthreads();

        // ---- exact min over this chunk for 8 cells per thread ------------
#pragma unroll
        for (int j = 0; j < 8; ++j) {
            const int cell = j * 512 + tid;
            const int yi = cell >> 6;
            const int xi = cell & 63;
            float m = minv[j];
            for (int n = 0; n < CHUNK; ++n)
                m = fminf(m, sDy[n][yi] + sDx[n][xi]);
            minv[j] = m;
        }

        // ---- WMMA: accumulate S tile (ty,tx), K stepped by 4 -------------
        // A (16x4 fp32): lanes 0-15 hold K=0,1 ; lanes 16-31 hold K=2,3.
        for (int k = 0; k < CHUNK; k += 4) {
            v2f a, bb;
            a.x  = sEy[k + 2 * kh + 0][ty * 16 + mi];
            a.y  = sEy[k + 2 * kh + 1][ty * 16 + mi];
            bb.x = sEx[k + 2 * kh + 0][tx * 16 + mi];
            bb.y = sEx[k + 2 * kh + 1][tx * 16 + mi];
            acc = __builtin_amdgcn_wmma_f32_16x16x4_f32(
                      false, a, false, bb, (short)0, acc, false, false);
        }
        __syncthreads();
    }

    // ---- stage S into LDS (reuse sDy); D layout: M = v + 8*kh ------------
    float (*S)[NC] = sDy;
#pragma unroll
    for (int v = 0; v < 8; ++v)
        S[ty * 16 + 8 * kh + v][tx * 16 + mi] = acc[v];
    __syncthreads();

    // ---- background row + reciprocal denominator -------------------------
    const float d = st_sizes[b] * 0.15f;
#pragma unroll
    for (int j = 0; j < 8; ++j) {
        const int cell = j * 512 + tid;
        const int yi = cell >> 6;
        const int xi = cell & 63;
        const float mind = fmaxf(minv[j], 0.0f);
        float bgd = d - sqrtf(mind);
        bgd = bgd * bgd;
        const float ebg = __expf(-bgd * INV2S2);
        const float inv = 1.0f / (S[yi][xi] + ebg);
        wsInv[b * CELLS + cell] = inv;
        out[((size_t)b * NROWS + 512) * CELLS + cell] = ebg * inv;  // bg row
    }
}

// ---------------------------------------------------------------------------
// Stage 2: stream out[b,n,cell] = Ey[n,yi] * Ex[n,xi] * inv[cell].
// 2048 blocks = 32 batches x 16 n-groups x 4 cell tiles; 256 threads;
// each thread owns 4 consecutive cells (float4) and loops 32 points.
// Non-temporal b128 stores: 256 MB write-once stream, fully coalesced.
// ---------------------------------------------------------------------------
__global__ __launch_bounds__(256)
void post_prob_stage2(const float* __restrict__ wsEy,
                      const float* __restrict__ wsEx,
                      const float* __restrict__ wsInv,
                      float* __restrict__ out) {
    const int tid  = threadIdx.x;
    const int tile = blockIdx.x & 3;          // cell tile (1024 cells)
    const int ng   = (blockIdx.x >> 2) & 15;  // point group (32 points)
    const int b    = blockIdx.x >> 6;

    const int cell0 = tile * 1024 + tid * 4;
    const int yi  = cell0 >> 6;
    const int xi0 = cell0 & 63;

    const float* Eyb = wsEy + (size_t)b * NPTS * NC;
    const float* Exb = wsEx + (size_t)b * NPTS * NC;
    const v4f iv = *(const v4f*)(wsInv + b * CELLS + cell0);
    float* ob = out + (size_t)b * NROWS * CELLS + cell0;

#pragma unroll 4
    for (int nn = 0; nn < 32; ++nn) {
        const int n = ng * 32 + nn;
        const float ey = Eyb[n * NC + yi];
        const v4f ex = *(const v4f*)(Exb + n * NC + xi0);
        const v4f r = ex * ey * iv;
        __builtin_nontemporal_store(r, (v4f*)(ob + (size_t)n * CELLS));
    }
}

// ---------------------------------------------------------------------------
extern "C" void kernel_launch(void* const* d_in, const int* in_sizes, int n_in,
                              void* d_out, int out_size, void* d_ws, size_t ws_size,
                              hipStream_t stream) {
    const float* points   = (const float*)d_in[0];  // (32, 512, 2)
    const float* st_sizes = (const float*)d_in[1];  // (32,)
    const float* cood     = (const float*)d_in[2];  // (64,)
    float* out = (float*)d_out;                     // (32, 513, 4096)

    // workspace layout: Ey | Ex | inv   (~8.5 MB total)
    float* wsEy  = (float*)d_ws;                        // 32*512*64
    float* wsEx  = wsEy + (size_t)32 * NPTS * NC;       // 32*512*64
    float* wsInv = wsEx + (size_t)32 * NPTS * NC;       // 32*4096

    post_prob_stage1<<<32, 512, 0, stream>>>(points, st_sizes, cood,
                                             wsEy, wsEx, wsInv, out);
    post_prob_stage2<<<2048, 256, 0, stream>>>(wsEy, wsEx, wsInv, out);

    (void)in_sizes; (void)n_in; (void)out_size; (void)ws_size;
}